// MeshUnpool_52261162058491
// MI455X (gfx1250) — compile-verified
//
#include <hip/hip_runtime.h>

// MeshUnpool for MI455X (gfx1250).
// B=8, C=128, E_OLD=48000, U=8000, E_NEW=72000.
// One 1024-thread workgroup per (b,c) row; build the full 72000-col output row
// in 288KB of LDS (CDNA5: 320KB/WGP) via coalesced b128 loads + LDS scatter,
// then stream it out with two TENSOR_STORE_FROM_LDS DMAs (one per CU-side TDM
// path of the WGP), each covering a contiguous 144KB half-row.

#define B_    8
#define C_    128
#define EOLD  48000
#define U_    8000
#define ENEW  (EOLD + 3 * U_)   // 72000

typedef unsigned int u32x4 __attribute__((ext_vector_type(4)));
typedef int          i32x8 __attribute__((ext_vector_type(8)));
typedef int          i32x4 __attribute__((ext_vector_type(4)));

#if defined(__gfx1250__) && __has_builtin(__builtin_amdgcn_tensor_store_from_lds)
// Issue one TDM store: contiguous `td0 x rows` f32 tile, LDS -> global.
// Descriptor packed per CDNA5 ISA §8.3/§8.4 (group0: count/lds_addr/57-bit
// global addr/type=2; group1: data_size=4B, dims/strides/tile dims).
__device__ __forceinline__ void tdm_store_f32(const float* gdst,
                                              unsigned ldsByteOff,
                                              unsigned td0, unsigned rows)
{
    const unsigned long long ga = (unsigned long long)(size_t)gdst;
    const unsigned ldsBase = __builtin_amdgcn_groupstaticsize() + ldsByteOff;

    u32x4 g0;
    g0[0] = 1u;                                    // count=1 (valid), user mode
    g0[1] = ldsBase;                               // lds_addr (bytes)
    g0[2] = (unsigned)ga;                          // global_addr[31:0]
    g0[3] = ((unsigned)(ga >> 32) & 0x01FFFFFFu)   // global_addr[56:32]
            | (2u << 30);                          // type = 2 ("image")

    i32x8 g1;
    g1[0] = (int)(2u << 16);                       // wg_mask=0, data_size=2 (4B)
    g1[1] = (int)((td0 & 0xFFFFu) << 16);          // tensor_dim0[15:0] @ bits63:48
    g1[2] = (int)((td0 >> 16) | ((rows & 0xFFFFu) << 16)); // tdim0 hi | tdim1 lo
    g1[3] = (int)((rows >> 16) | (td0 << 16));     // tdim1 hi | tile_dim0
    g1[4] = (int)rows;                             // tile_dim1=rows, tile_dim2=0
    g1[5] = (int)td0;                              // tensor_dim0_stride[31:0]
    g1[6] = 0;                                     // stride0 hi | stride1 lo
    g1[7] = 0;                                     // stride1 hi

    i32x4 gz4 = {0, 0, 0, 0};                      // groups 2/3 unused (2D)
    i32x8 gz8 = {0, 0, 0, 0, 0, 0, 0, 0};          // VADDR4 group: unused

    __builtin_amdgcn_tensor_store_from_lds(g0, g1, gz4, gz4, gz8, 0);
}
#endif

extern "C" __global__ __launch_bounds__(1024)
void mesh_unpool_kernel(const float* __restrict__ features,   // (B,C,EOLD)
                        const float* __restrict__ weights,    // (B,U,2,3)
                        const int*   __restrict__ old_idx,    // (B,EOLD)
                        const int*   __restrict__ new_idx,    // (B,U)
                        const int*   __restrict__ new_left,   // (B,U)
                        const int*   __restrict__ new_right,  // (B,U)
                        float* __restrict__ out)              // (B,C,ENEW)
{
    extern __shared__ float lds[];   // ENEW floats = 288000 bytes

    const int tid = (int)threadIdx.x;
    const int bc  = (int)blockIdx.x;        // b*C + c  (consecutive c share idx arrays in L2)
    const int b   = bc / C_;

    const float* __restrict__ fRow = features + (size_t)bc * EOLD;
    const int*   __restrict__ oi   = old_idx  + (size_t)b * EOLD;
    const float* __restrict__ wB   = weights  + (size_t)b * U_ * 6;  // [u][s][k] -> u*6 + s*3 + k
    const int*   __restrict__ nI   = new_idx   + (size_t)b * U_;
    const int*   __restrict__ nL   = new_left  + (size_t)b * U_;
    const int*   __restrict__ nR   = new_right + (size_t)b * U_;

    // ---- Phase A: scatter (scaled) old columns into LDS ---------------------
    // column p < U  -> scale lw0 = w[p][0][0];  U<=p<2U -> rw0 = w[p-U][1][0];
    // p >= 2U -> scale 1.  Region boundaries (U, 2U) are multiples of 4, so a
    // float4 group never straddles regions.
    for (int p4 = tid; p4 < EOLD / 4; p4 += 1024) {
        const int    p   = p4 * 4;
        const float4 f   = *(const float4*)(fRow + p);
        const int4   idx = *(const int4*)(oi + p);
        float s0 = 1.f, s1 = 1.f, s2 = 1.f, s3 = 1.f;
        if (p < U_) {
            s0 = wB[(p + 0) * 6 + 0]; s1 = wB[(p + 1) * 6 + 0];
            s2 = wB[(p + 2) * 6 + 0]; s3 = wB[(p + 3) * 6 + 0];
        } else if (p < 2 * U_) {
            const int q = p - U_;
            s0 = wB[(q + 0) * 6 + 3]; s1 = wB[(q + 1) * 6 + 3];
            s2 = wB[(q + 2) * 6 + 3]; s3 = wB[(q + 3) * 6 + 3];
        }
        lds[idx.x] = f.x * s0;
        lds[idx.y] = f.y * s1;
        lds[idx.z] = f.z * s2;
        lds[idx.w] = f.w * s3;
    }

    // ---- Phase B: the 3*U new columns --------------------------------------
    // lf[q] = features[:,q], rf[q] = features[:,U+q]
    // out[new_left[q]]  = lf*w[q][0][1]
    // out[new_right[q]] = rf*w[q][1][1]
    // out[new_idx[q]]   = 0.5*(lf*w[q][0][2] + rf*w[q][1][2])
    for (int q4 = tid; q4 < U_ / 4; q4 += 1024) {
        const int    q  = q4 * 4;
        const float4 lf = *(const float4*)(fRow + q);
        const float4 rf = *(const float4*)(fRow + U_ + q);
        const int4   ni = *(const int4*)(nI + q);
        const int4   nl = *(const int4*)(nL + q);
        const int4   nr = *(const int4*)(nR + q);
        const float lfv[4] = {lf.x, lf.y, lf.z, lf.w};
        const float rfv[4] = {rf.x, rf.y, rf.z, rf.w};
        const int   niv[4] = {ni.x, ni.y, ni.z, ni.w};
        const int   nlv[4] = {nl.x, nl.y, nl.z, nl.w};
        const int   nrv[4] = {nr.x, nr.y, nr.z, nr.w};
#pragma unroll
        for (int j = 0; j < 4; ++j) {
            const float* w = wB + (q + j) * 6;
            lds[nlv[j]] = lfv[j] * w[1];
            lds[nrv[j]] = rfv[j] * w[4];
            lds[niv[j]] = 0.5f * (lfv[j] * w[2] + rfv[j] * w[5]);
        }
    }

    __syncthreads();   // full output row now resident in LDS

    // ---- Phase C: stream LDS row -> global ---------------------------------
#if defined(__gfx1250__) && __has_builtin(__builtin_amdgcn_tensor_store_from_lds)
    {
        // Two DMAs, one issued from lane 0 of wave 0 and of wave 1, each a
        // contiguous 36000-element (144000 B) half: tile 7200 x 5 of f32.
        const int wave = tid >> 5;
        if ((tid & 31) == 0 && wave < 2) {
            const unsigned halfElems = ENEW / 2;       // 36000
            const unsigned td0 = 7200u, rows = 5u;     // 7200*5 = 36000
            tdm_store_f32(out + (size_t)bc * ENEW + (size_t)wave * halfElems,
                          wave * (halfElems * 4u), td0, rows);
            __builtin_amdgcn_s_wait_tensorcnt(0);      // hold LDS until DMA done
        }
    }
#else
    {
        float* __restrict__ oRow = out + (size_t)bc * ENEW;
        for (int e4 = tid; e4 < ENEW / 4; e4 += 1024) {
            *(float4*)(oRow + e4 * 4) = *(const float4*)(lds + e4 * 4);
        }
    }
#endif
}

extern "C" void kernel_launch(void* const* d_in, const int* in_sizes, int n_in,
                              void* d_out, int out_size, void* d_ws, size_t ws_size,
                              hipStream_t stream) {
    // setup_inputs order:
    // 0 features, 1 weights, 2 old_idx, 3 left_idx, 4 right_idx,
    // 5 new_idx, 6 new_left_idx, 7 new_right_idx
    const float* features = (const float*)d_in[0];
    const float* weights  = (const float*)d_in[1];
    const int*   old_idx  = (const int*)d_in[2];
    const int*   new_idx  = (const int*)d_in[5];
    const int*   new_left = (const int*)d_in[6];
    const int*   new_right= (const int*)d_in[7];
    float* out = (float*)d_out;

    const dim3 grid(B_ * C_);
    const dim3 block(1024);
    const size_t shmem = (size_t)ENEW * sizeof(float);   // 288000 B < 320KB/WGP

    hipLaunchKernelGGL(mesh_unpool_kernel, grid, block, shmem, stream,
                       features, weights, old_idx, new_idx, new_left, new_right, out);
}